// get_loss_25434796327294
// MI455X (gfx1250) — compile-verified
//
#include <hip/hip_runtime.h>
#include <hip/hip_bf16.h>

typedef __attribute__((ext_vector_type(2))) float v2f;
typedef __attribute__((ext_vector_type(8))) float v8f;

#define BSZ 4
#define HH 64
#define WW 64
#define HW (HH * WW)       // 4096
#define NG 2
#define NP 192
#define STEP 20
#define NV (HW / STEP)     // 204
#define NTILE (NP / 16)    // 12
#define NBLK (BSZ * NG * NV)  // 1632
#define SCORING_WEIGHT 0.01f

// ---------------------------------------------------------------------------
// Kernel A: T[b,g,i,p] = gt_r[b]·grps[g] + gt_t[b];  t2[b,g,p] = sum_i T^2
// ---------------------------------------------------------------------------
__global__ void __launch_bounds__(192)
precompute_T_kernel(const float* __restrict__ gt_r,
                    const float* __restrict__ gt_t,
                    const float* __restrict__ grps,
                    float* __restrict__ Tws,    // [BSZ][NG][3][NP]
                    float* __restrict__ t2ws)   // [BSZ][NG][NP]
{
    int bg = blockIdx.x;            // b*NG + g
    int b  = bg / NG;
    int g  = bg % NG;
    int p  = threadIdx.x;
    if (p >= NP) return;

    const float* Rb = gt_r + b * 9;
    const float* tb = gt_t + b * 3;
    const float* G  = grps + g * 3 * NP;
    float g0 = G[0 * NP + p], g1 = G[1 * NP + p], g2 = G[2 * NP + p];

    float* Tout = Tws + bg * 3 * NP;
    float t2 = 0.0f;
#pragma unroll
    for (int i = 0; i < 3; ++i) {
        float v = Rb[i * 3 + 0] * g0 + Rb[i * 3 + 1] * g1 + Rb[i * 3 + 2] * g2 + tb[i];
        Tout[i * NP + p] = v;
        t2 += v * v;
    }
    t2ws[bg * NP + p] = t2;
}

// ---------------------------------------------------------------------------
// Kernel B: per (b,g,n) build P, run 12x12 WMMA tiles of cross = T^T * P,
// min over p, mean over t, write scalar partial contribution.
// ---------------------------------------------------------------------------
__global__ void __launch_bounds__(128)
chamfer_wmma_kernel(const float* __restrict__ pred_r,   // [BSZ][4][HW]
                    const float* __restrict__ pred_t,   // [BSZ][3][HW]
                    const float* __restrict__ pred_s,   // [BSZ][1][HW]
                    const float* __restrict__ grps,     // [NG][3][NP]
                    const float* __restrict__ Tws,
                    const float* __restrict__ t2ws,
                    float* __restrict__ partial)        // [NBLK]
{
    __shared__ float Tsh[3][NP];
    __shared__ float Psh[3][NP];
    __shared__ float t2sh[NP];
    __shared__ float p2sh[NP];
    __shared__ float wavesum[4];

    const int blk = blockIdx.x;     // b*NG*NV + g*NV + n
    const int n   = blk % NV;
    const int bg  = blk / NV;
    const int g   = bg % NG;
    const int b   = bg / NG;
    const int pix = n * STEP;
    const int tid = threadIdx.x;

    // Quaternion -> rotation (uniform scalar loads, replicated per thread)
    float q0 = pred_r[(b * 4 + 0) * HW + pix];
    float q1 = pred_r[(b * 4 + 1) * HW + pix];
    float q2 = pred_r[(b * 4 + 2) * HW + pix];
    float q3 = pred_r[(b * 4 + 3) * HW + pix];
    float inv = 1.0f / sqrtf(q0 * q0 + q1 * q1 + q2 * q2 + q3 * q3);
    q0 *= inv; q1 *= inv; q2 *= inv; q3 *= inv;

    float R00 = 1.0f - 2.0f * (q2 * q2 + q3 * q3);
    float R01 = 2.0f * q1 * q2 - 2.0f * q0 * q3;
    float R02 = 2.0f * q0 * q2 + 2.0f * q1 * q3;
    float R10 = 2.0f * q1 * q2 + 2.0f * q3 * q0;
    float R11 = 1.0f - 2.0f * (q1 * q1 + q3 * q3);
    float R12 = -2.0f * q0 * q1 + 2.0f * q2 * q3;
    float R20 = -2.0f * q0 * q2 + 2.0f * q1 * q3;
    float R21 = 2.0f * q0 * q1 + 2.0f * q2 * q3;
    float R22 = 1.0f - 2.0f * (q1 * q1 + q2 * q2);

    float tx = pred_t[(b * 3 + 0) * HW + pix];
    float ty = pred_t[(b * 3 + 1) * HW + pix];
    float tz = pred_t[(b * 3 + 2) * HW + pix];

    // Stage T / t2 (precomputed) and build P / p2 into LDS
    const float* G2  = grps + g * 3 * NP;
    const float* Tg  = Tws + bg * 3 * NP;
    const float* t2g = t2ws + bg * NP;
    for (int p = tid; p < NP; p += 128) {
        float g0v = G2[0 * NP + p], g1v = G2[1 * NP + p], g2v = G2[2 * NP + p];
        float P0 = R00 * g0v + R01 * g1v + R02 * g2v + tx;
        float P1 = R10 * g0v + R11 * g1v + R12 * g2v + ty;
        float P2 = R20 * g0v + R21 * g1v + R22 * g2v + tz;
        Psh[0][p] = P0; Psh[1][p] = P1; Psh[2][p] = P2;
        p2sh[p] = P0 * P0 + P1 * P1 + P2 * P2;
        Tsh[0][p] = Tg[0 * NP + p];
        Tsh[1][p] = Tg[1 * NP + p];
        Tsh[2][p] = Tg[2 * NP + p];
        t2sh[p] = t2g[p];
    }
    __syncthreads();

    const int lane = tid & 31;
    const int wave = tid >> 5;
    const int half = lane >> 4;     // 0: K rows 0/1, 1: K rows 2/3(pad)
    const int m    = lane & 15;

    float tsum = 0.0f;              // sum over this wave's T points of min-dist

    // Each wave: 3 T-tiles (tt = wave, wave+4, wave+8), 12 P-tiles each.
    for (int tt = wave; tt < NTILE; tt += 4) {
        const int t0 = tt * 16;

        // A fragment: T^T tile, 16(M=t) x 4(K=i), K=3 zero-padded
        v2f afrag;
        afrag.x = half ? Tsh[2][t0 + m] : Tsh[0][t0 + m];
        afrag.y = half ? 0.0f          : Tsh[1][t0 + m];

        float t2v[8];
#pragma unroll
        for (int r = 0; r < 8; ++r) t2v[r] = t2sh[t0 + r + half * 8];

        float minv[8];
#pragma unroll
        for (int r = 0; r < 8; ++r) minv[r] = 3.4e38f;

        for (int pt_ = 0; pt_ < NTILE; ++pt_) {
            const int p0 = pt_ * 16;
            // B fragment: P tile, 4(K=i) x 16(N=p), K=3 zero-padded
            v2f bfrag;
            bfrag.x = half ? Psh[2][p0 + m] : Psh[0][p0 + m];
            bfrag.y = half ? 0.0f          : Psh[1][p0 + m];

            v8f c = {};
            // cross[t,p] = sum_i T[i,t] * P[i,p]   (one WMMA, K=4>=3)
            c = __builtin_amdgcn_wmma_f32_16x16x4_f32(
                    false, afrag, false, bfrag, (short)0, c, false, false);

            const float p2v = p2sh[p0 + m];
#pragma unroll
            for (int r = 0; r < 8; ++r) {
                float val = t2v[r] + p2v - 2.0f * c[r];
                minv[r] = fminf(minv[r], val);
            }
        }

        // min across the 16 lanes holding the same t (sqrt/clamp are monotone,
        // so reduce the quadratic form first, one sqrt per t)
#pragma unroll
        for (int r = 0; r < 8; ++r) {
            float mv = minv[r];
            mv = fminf(mv, __shfl_xor(mv, 1, 32));
            mv = fminf(mv, __shfl_xor(mv, 2, 32));
            mv = fminf(mv, __shfl_xor(mv, 4, 32));
            mv = fminf(mv, __shfl_xor(mv, 8, 32));
            float d = sqrtf(fmaxf(mv, 1e-12f));
            if (m == 0) tsum += d;   // lanes 0 and 16 each own 8 t values
        }
    }

    // wave reduction of tsum
    tsum += __shfl_xor(tsum, 1, 32);
    tsum += __shfl_xor(tsum, 2, 32);
    tsum += __shfl_xor(tsum, 4, 32);
    tsum += __shfl_xor(tsum, 8, 32);
    tsum += __shfl_xor(tsum, 16, 32);
    if (lane == 0) wavesum[wave] = tsum;
    __syncthreads();

    if (tid == 0) {
        float s = wavesum[0] + wavesum[1] + wavesum[2] + wavesum[3];
        float add_ij = s * (1.0f / (float)NP);            // mean over t
        float ps = pred_s[b * HW + pix];
        float contrib = add_ij * ps * (1.0f / (float)NG); // mean over g (split)
        if (g == 0) contrib -= SCORING_WEIGHT * logf(ps); // count log term once
        contrib *= 1.0f / ((float)BSZ * (float)NV);       // mean over n and b
        partial[blk] = contrib;
    }
}

// ---------------------------------------------------------------------------
// Kernel C: deterministic fixed-order reduction of per-block partials
// ---------------------------------------------------------------------------
__global__ void __launch_bounds__(256)
reduce_kernel(const float* __restrict__ partial, float* __restrict__ out)
{
    __shared__ float sh[256];
    int tid = threadIdx.x;
    float s = 0.0f;
    for (int i = tid; i < NBLK; i += 256) s += partial[i];
    sh[tid] = s;
    __syncthreads();
    for (int off = 128; off > 0; off >>= 1) {
        if (tid < off) sh[tid] += sh[tid + off];
        __syncthreads();
    }
    if (tid == 0) out[0] = sh[0];
}

// ---------------------------------------------------------------------------
extern "C" void kernel_launch(void* const* d_in, const int* in_sizes, int n_in,
                              void* d_out, int out_size, void* d_ws, size_t ws_size,
                              hipStream_t stream)
{
    const float* pred_r = (const float*)d_in[0];
    const float* pred_t = (const float*)d_in[1];
    const float* pred_s = (const float*)d_in[2];
    const float* gt_r   = (const float*)d_in[3];
    const float* gt_t   = (const float*)d_in[4];
    const float* grps   = (const float*)d_in[5];
    // d_in[6] mask, d_in[7] cls_ids: unused by the reference
    // d_in[8] step: fixed at 20 (compile-time, like BS/H/W/NG/NP)
    float* out = (float*)d_out;

    float* Tws     = (float*)d_ws;                       // BSZ*NG*3*NP
    float* t2ws    = Tws + BSZ * NG * 3 * NP;            // BSZ*NG*NP
    float* partial = t2ws + BSZ * NG * NP;               // NBLK

    precompute_T_kernel<<<BSZ * NG, 192, 0, stream>>>(gt_r, gt_t, grps, Tws, t2ws);
    chamfer_wmma_kernel<<<NBLK, 128, 0, stream>>>(pred_r, pred_t, pred_s, grps,
                                                  Tws, t2ws, partial);
    reduce_kernel<<<1, 256, 0, stream>>>(partial, out);
}